// FFEdgeCountingLayer_90443421319695
// MI455X (gfx1250) — compile-verified
//
#include <hip/hip_runtime.h>
#include <hip/hip_bf16.h>
#include <math.h>
#include <stdint.h>

// Problem constants (from reference): OPS=2 operator slots, E=3 edge types.
static constexpr int kOPS = 2;
static constexpr int kE   = 3;

// Main-kernel tiling: 32 n  x 32 o per block, K-chunk of 64, 16x16 threads.
static constexpr int TNF = 32;          // n-tile
static constexpr int TOF = 32;          // o-tile
static constexpr int KC  = 64;          // k-chunk (floats)
static constexpr int SXS = KC + 4;      // x-tile LDS row stride (16B-aligned pad)

// ---------------------------------------------------------------------------
// gfx1250 async global->LDS copy (ASYNCcnt path).  One asm statement issues a
// 16-byte transfer per active lane, from the global address in %1 to the LDS address in %0.
// ---------------------------------------------------------------------------
__device__ __forceinline__ void async_ld16(uint32_t lds_byte_addr, const void* gptr) {
  asm volatile("global_load_async_to_lds_b128 %0, %1, off"
               :
               : "v"(lds_byte_addr), "v"((unsigned long long)(uintptr_t)gptr)
               : "memory");
}
__device__ __forceinline__ void wait_async0() {
  asm volatile("s_wait_asynccnt 0" ::: "memory");
}

// ---------------------------------------------------------------------------
// Phase 1a: per-output operator selection (hard Gumbel argmax over OPS=2).
// sgn[o] = +1 for t-norm (min-reduce), -1 for t-conorm (max-reduce).
// ---------------------------------------------------------------------------
__global__ void op_select_kernel(const float* __restrict__ op_logits,
                                 const float* __restrict__ u_op,
                                 int* __restrict__ op_idx,
                                 float* __restrict__ sgn,
                                 int O) {
  int o = blockIdx.x * blockDim.x + threadIdx.x;
  if (o >= O) return;
  float z0 = op_logits[o * kOPS + 0] + (-logf(-logf(u_op[o * kOPS + 0])));
  float z1 = op_logits[o * kOPS + 1] + (-logf(-logf(u_op[o * kOPS + 1])));
  float m  = fmaxf(z0, z1);
  float e0 = expf(z0 - m), e1 = expf(z1 - m);
  float s  = e0 + e1;
  float y0 = e0 / s, y1 = e1 / s;
  int idx  = (y1 > y0) ? 1 : 0;     // first-max semantics like jnp.argmax
  op_idx[o] = idx;
  sgn[o]    = (idx == 0) ? 1.0f : -1.0f;
}

// ---------------------------------------------------------------------------
// Phase 1b: per-(o,i) straight-through edge weights for the chosen operator.
// Packs float4{ sel_identity, sel_complement, sel_noedge * no_edge_offset, 0 }.
// Non-selected slots are exactly 0 (ys - ys == 0), selected slot is
// (1+ys)-ys, matching the reference's forward values.
// ---------------------------------------------------------------------------
__global__ void edge_select_kernel(const float* __restrict__ edge_logits,
                                   const float* __restrict__ u_edge,
                                   const int* __restrict__ op_idx,
                                   float4* __restrict__ abc,
                                   int O, int I) {
  int t = blockIdx.x * blockDim.x + threadIdx.x;
  if (t >= O * I) return;
  int o = t / I;
  int i = t - o * I;
  int p = op_idx[o];
  size_t base = (((size_t)o * kOPS + p) * I + i) * kE;

  float z[kE];
  float m = -INFINITY;
#pragma unroll
  for (int e = 0; e < kE; ++e) {
    float g = -logf(-logf(u_edge[base + e]));
    z[e] = edge_logits[base + e] + g;
    m = fmaxf(m, z[e]);
  }
  float ys[kE];
  float s = 0.0f;
#pragma unroll
  for (int e = 0; e < kE; ++e) { ys[e] = expf(z[e] - m); s += ys[e]; }
#pragma unroll
  for (int e = 0; e < kE; ++e) ys[e] = ys[e] / s;

  int idx = 0;                       // first occurrence of max, on y_soft
  if (ys[1] > ys[idx]) idx = 1;
  if (ys[2] > ys[idx]) idx = 2;

  float sel[kE];
#pragma unroll
  for (int e = 0; e < kE; ++e)
    sel[e] = (((e == idx) ? 1.0f : 0.0f) + ys[e]) - ys[e];  // straight-through

  float off = (p == 0) ? 1.0f : 0.0f;   // no_edge offset: 1 for t-norm, 0 for t-conorm
  float4 q;
  q.x = sel[0];
  q.y = sel[1];
  q.z = sel[2] * off;
  q.w = 0.0f;
  abc[(size_t)o * I + i] = q;
}

// ---------------------------------------------------------------------------
// Phase 2: out[n,o] = reduce_i max( a*x , b*(1-x) , c )   (min if t-norm).
// Tropical (mul/max -> min/max) reduction: not WMMA-expressible; done as a
// register-blocked wave32 VALU kernel with async global->LDS staging.
// Assumes N%TNF==0, O%TOF==0, I%KC==0 (holds: 1024/256/512).
// ---------------------------------------------------------------------------
__global__ __launch_bounds__(256)
void ff_logic_main(const float* __restrict__ x,
                   const float4* __restrict__ abc,
                   const float* __restrict__ sgn,
                   float* __restrict__ out,
                   int N, int O, int I) {
  __shared__ float  sx[TNF * SXS];       // x tile, padded rows
  __shared__ float4 sabc[TOF * KC];      // edge-weight tile

  const int tx  = threadIdx.x;           // 0..15
  const int ty  = threadIdx.y;           // 0..15
  const int tid = ty * 16 + tx;          // 0..255
  const int n0  = blockIdx.x * TNF;
  const int o0  = blockIdx.y * TOF;

  // LDS logical byte addresses (low 32 bits of the flat shared-aperture addr).
  const uint32_t sx_base   = (uint32_t)(uintptr_t)(&sx[0]);
  const uint32_t sabc_base = (uint32_t)(uintptr_t)(&sabc[0]);

  const float sg0 = sgn[o0 + 2 * ty + 0];
  const float sg1 = sgn[o0 + 2 * ty + 1];

  float acc00 = INFINITY, acc01 = INFINITY;   // acc{o-sub}{n-sub}, min over sgn*ev
  float acc10 = INFINITY, acc11 = INFINITY;

  for (int k0 = 0; k0 < I; k0 += KC) {
    __syncthreads();  // previous chunk's LDS reads complete before overwrite

    // Stage x tile: TNF rows x KC floats = 512 x 16B transfers, 2 per thread.
#pragma unroll
    for (int j = 0; j < 2; ++j) {
      int v   = tid * 2 + j;           // 0..511
      int row = v >> 4;                // /16  -> n row in tile
      int kq  = (v & 15) << 2;         // *4   -> float column
      const float* gp = x + (size_t)(n0 + row) * I + k0 + kq;
      async_ld16(sx_base + (uint32_t)(row * SXS + kq) * 4u, gp);
    }
    // Stage abc tile: TOF rows x KC float4 = 2048 x 16B transfers, 8/thread.
#pragma unroll
    for (int j = 0; j < 8; ++j) {
      int v   = tid * 8 + j;           // 0..2047
      int row = v >> 6;                // /64 -> o row in tile
      int kq  = v & 63;                // float4 column
      const float4* gp = abc + (size_t)(o0 + row) * I + k0 + kq;
      async_ld16(sabc_base + (uint32_t)(row * KC + kq) * 16u, gp);
    }
    wait_async0();     // this wave's async transfers landed in LDS
    __syncthreads();   // all waves' tiles visible

    const float*  xr0 = &sx[(2 * tx + 0) * SXS];
    const float*  xr1 = &sx[(2 * tx + 1) * SXS];
    const float4* q0r = &sabc[(2 * ty + 0) * KC];
    const float4* q1r = &sabc[(2 * ty + 1) * KC];

#pragma unroll 8
    for (int kk = 0; kk < KC; ++kk) {
      float xv0 = xr0[kk], xv1 = xr1[kk];
      float xc0 = 1.0f - xv0, xc1 = 1.0f - xv1;
      float4 q0 = q0r[kk];
      float4 q1 = q1r[kk];
      float ev;
      ev = fmaxf(fmaxf(q0.x * xv0, q0.y * xc0), q0.z); acc00 = fminf(acc00, sg0 * ev);
      ev = fmaxf(fmaxf(q0.x * xv1, q0.y * xc1), q0.z); acc01 = fminf(acc01, sg0 * ev);
      ev = fmaxf(fmaxf(q1.x * xv0, q1.y * xc0), q1.z); acc10 = fminf(acc10, sg1 * ev);
      ev = fmaxf(fmaxf(q1.x * xv1, q1.y * xc1), q1.z); acc11 = fminf(acc11, sg1 * ev);
    }
  }

  const int n_a = n0 + 2 * tx;
  const int o_a = o0 + 2 * ty;
  out[(size_t)(n_a + 0) * O + (o_a + 0)] = sg0 * acc00;
  out[(size_t)(n_a + 1) * O + (o_a + 0)] = sg0 * acc01;
  out[(size_t)(n_a + 0) * O + (o_a + 1)] = sg1 * acc10;
  out[(size_t)(n_a + 1) * O + (o_a + 1)] = sg1 * acc11;
}

// ---------------------------------------------------------------------------
extern "C" void kernel_launch(void* const* d_in, const int* in_sizes, int n_in,
                              void* d_out, int out_size, void* d_ws, size_t ws_size,
                              hipStream_t stream) {
  const float* x           = (const float*)d_in[0];  // [N, I]
  const float* edge_logits = (const float*)d_in[1];  // [O, 2, I, 3]
  const float* op_logits   = (const float*)d_in[2];  // [O, 2]
  const float* u_edge      = (const float*)d_in[3];  // [O, 2, I, 3]
  const float* u_op        = (const float*)d_in[4];  // [O, 2]

  const int O = in_sizes[2] / kOPS;
  const int I = in_sizes[1] / (O * kOPS * kE);
  const int N = in_sizes[0] / I;

  // Workspace layout: [float4 abc: O*I][int op_idx: O][float sgn: O]
  float4* abc = (float4*)d_ws;
  int*    opi = (int*)((char*)d_ws + (size_t)O * I * sizeof(float4));
  float*  sg  = (float*)(opi + O);

  op_select_kernel<<<(O + 255) / 256, 256, 0, stream>>>(op_logits, u_op, opi, sg, O);

  const int tot = O * I;
  edge_select_kernel<<<(tot + 255) / 256, 256, 0, stream>>>(edge_logits, u_edge, opi,
                                                            abc, O, I);

  dim3 grid(N / TNF, O / TOF);
  dim3 block(16, 16);
  ff_logic_main<<<grid, block, 0, stream>>>(x, abc, sg, (float*)d_out, N, O, I);
}